// SecondGeometry_13331578487371
// MI455X (gfx1250) — compile-verified
//
#include <hip/hip_runtime.h>
#include <math.h>
#include <stdint.h>

typedef float v2f __attribute__((ext_vector_type(2)));
typedef float v8f __attribute__((ext_vector_type(8)));

#define DD 24
#define HH 512
#define HHALF 256         // H-range per wave (2 waves share one row-group)
#define LDW 26            // padded LDS row stride (floats): conflict-free, 8B-aligned rows
#define NGROUPS 4         // row-groups per block (16 rows each)
#define ROWS_PER_BLOCK 64 // 4 groups x 16 rows; 8 waves (2 per group)

// Half-wave swap (lane l <-> l^16) via DS_SWIZZLE SWAPX16: xor=0x10, and=0x1f.
__device__ __forceinline__ float swap16f(float v) {
  return __int_as_float(__builtin_amdgcn_ds_swizzle(__float_as_int(v), 0x401f));
}

// CDNA5 hardware V_TANH_F32 TRANS op; co-executes with the WMMA pipe.
__device__ __forceinline__ float fast_tanh(float u) {
#if __has_builtin(__builtin_amdgcn_tanhf)
  return __builtin_amdgcn_tanhf(u);
#elif __has_builtin(__builtin_amdgcn_tanh_f32)
  return __builtin_amdgcn_tanh_f32(u);
#else
  return tanhf(u);
#endif
}

// acc = -f + g * ((g.f) - v^T H v) / (1 + g.g)
//   g = W1^T (W2 * (1-t^2)),  t = tanh(W1 p + b1)
//   v^T H v = sum_n W2_n * (-2 t_n (1-t_n^2)) * (W1 v)_n^2
//   f = ks*(p - attractor) + kd*v
__global__ __launch_bounds__(256) void secgeo_wmma_kernel(
    const float* __restrict__ x,
    const float* __restrict__ attractor,
    const float* __restrict__ W1,
    const float* __restrict__ b1,
    const float* __restrict__ W2,
    const float* __restrict__ kstiff,
    const float* __restrict__ kdiss,
    float* __restrict__ out)
{
  __shared__ float w1s[HH * LDW + 8];
  __shared__ float b1s[HH];
  __shared__ float w2s[HH];
  __shared__ float glsP[NGROUPS][16][24];  // partner-partial then combined g
  __shared__ float qlsP[NGROUPS][16];      // partner-partial v^T H v

  const int tid = threadIdx.x;

  // Stage W1 (512x24) into stride-26 LDS rows via async global->LDS copies
  // (ASYNCcnt path: data never touches VGPRs). 12 B64 chunks per row, all
  // 8B-aligned on both sides.
  for (int r = tid; r < HH; r += 256) {
    const float* gsrc = W1 + r * DD;
    const uint32_t ldst = (uint32_t)(uintptr_t)(w1s + r * LDW);
#pragma unroll
    for (int c = 0; c < DD; c += 2) {
      const float* gaddr = gsrc + c;
      const uint32_t laddr = ldst + (uint32_t)(c * 4);
      asm volatile("global_load_async_to_lds_b64 %0, %1, off"
                   :: "v"(laddr), "v"(gaddr) : "memory");
    }
    b1s[r] = b1[r];
    w2s[r] = W2[r];
  }
  asm volatile("s_wait_asynccnt 0x0" ::: "memory");
  __syncthreads();

  const int lane  = tid & 31;
  const int wave  = tid >> 5;
  const int group = wave >> 1;       // row-group (16 batch rows)
  const int wh    = wave & 1;        // which half of H this wave reduces
  const int col   = lane & 15;       // N index (batch row within tile)
  const int hi    = (lane >> 4) & 1; // lane half
  const int b0    = blockIdx.x * ROWS_PER_BLOCK + group * 16;
  const int brow  = b0 + col;

  // Preload B operands (f32 B-layout, K=i): P^T and V^T columns for the group's 16 rows.
  v2f pB[6], vB[6];
  const float* xr = x + brow * (2 * DD);
#pragma unroll
  for (int s = 0; s < 6; ++s) {
    const int i0 = 4 * s + 2 * hi;
    pB[s] = *(const v2f*)(xr + i0);
    vB[s] = *(const v2f*)(xr + DD + i0);
  }

  v8f cg0 = {};  // G^T tile, i = 0..15
  v8f cg1 = {};  // G^T tile, i = 16..31 (24..31 garbage, never read)
  float qacc = 0.f;

  const int nbBeg = wh * HHALF;
  const int nbEnd = nbBeg + HHALF;
  for (int nb = nbBeg; nb < nbEnd; nb += 16) {
    // GEMM1/2: C[n,b] += W1[n,i] * {P,V}[b,i]  (A = W1 rows from LDS, 6 K-steps of 4)
    const float* arow = w1s + (nb + col) * LDW + 2 * hi;
    v8f cu = {}, cw = {};
#pragma unroll
    for (int s = 0; s < 6; ++s) {
      v2f a = *(const v2f*)(arow + 4 * s);
      cu = __builtin_amdgcn_wmma_f32_16x16x4_f32(false, a, false, pB[s], (short)0, cu, false, false);
      cw = __builtin_amdgcn_wmma_f32_16x16x4_f32(false, a, false, vB[s], (short)0, cw, false, false);
    }

    // Elementwise: a_n = W2*(1-t^2); qacc += W2*(-2 t (1-t^2)) * (W1 v)_n^2
    float av[8];
#pragma unroll
    for (int q = 0; q < 8; ++q) {
      const int n = nb + q + 8 * hi;
      const float u   = cu[q] + b1s[n];
      const float t   = fast_tanh(u);
      const float om  = 1.f - t * t;
      const float w2v = w2s[n];
      av[q] = w2v * om;
      const float wv = cw[q];
      qacc += w2v * (-2.f * t * om) * (wv * wv);
    }

    // C-layout (n,b) -> B-layout (k=n, N=b): half-wave swap + select, no LDS round-trip.
    float sw[8];
#pragma unroll
    for (int q = 0; q < 8; ++q) sw[q] = swap16f(av[q]);

    v2f Bk[4];
    Bk[0].x = hi ? sw[2] : av[0];  Bk[0].y = hi ? sw[3] : av[1];  // k0=0
    Bk[1].x = hi ? sw[6] : av[4];  Bk[1].y = hi ? sw[7] : av[5];  // k0=4
    Bk[2].x = hi ? av[2] : sw[0];  Bk[2].y = hi ? av[3] : sw[1];  // k0=8
    Bk[3].x = hi ? av[6] : sw[4];  Bk[3].y = hi ? av[7] : sw[5];  // k0=12

    // GEMM3: G^T[i,b] += W1[n,i] * a[n,b], A = W1 transposed reads from LDS.
#pragma unroll
    for (int k = 0; k < 4; ++k) {
      const int rk = nb + 4 * k + 2 * hi;
      const float* r0 = w1s + rk * LDW;
      const float* r1 = w1s + (rk + 1) * LDW;
      v2f a0; a0.x = r0[col];      a0.y = r1[col];
      v2f a1; a1.x = r0[16 + col]; a1.y = r1[16 + col];
      cg0 = __builtin_amdgcn_wmma_f32_16x16x4_f32(false, a0, false, Bk[k], (short)0, cg0, false, false);
      cg1 = __builtin_amdgcn_wmma_f32_16x16x4_f32(false, a1, false, Bk[k], (short)0, cg1, false, false);
    }
  }

  // Partial v^T H v for this wave's H-half (both lane halves hold disjoint n-subsets).
  float qv = qacc + swap16f(qacc);

  // H-split combine: odd wave publishes partials, even wave reduces + finishes.
  if (wh == 1) {
#pragma unroll
    for (int q = 0; q < 8; ++q) glsP[group][col][q + 8 * hi] = cg0[q];
    if (hi == 0) {
#pragma unroll
      for (int q = 0; q < 8; ++q) glsP[group][col][16 + q] = cg1[q];
    }
    if (lane < 16) qlsP[group][lane] = qv;
  }
  __syncthreads();

  if (wh == 0) {
    qv += qlsP[group][col];
#pragma unroll
    for (int q = 0; q < 8; ++q) cg0[q] += glsP[group][col][q + 8 * hi];
    if (hi == 0) {
#pragma unroll
      for (int q = 0; q < 8; ++q) cg1[q] += glsP[group][col][16 + q];
    }

    // Stage combined g = G[b, 0..23] (same-wave DS ordering: read-then-overwrite is safe).
#pragma unroll
    for (int q = 0; q < 8; ++q) glsP[group][col][q + 8 * hi] = cg0[q];
    if (hi == 0) {
#pragma unroll
      for (int q = 0; q < 8; ++q) glsP[group][col][16 + q] = cg1[q];
    }

    // Finale: one batch row per lane (lanes 0..15).
    if (lane < 16) {
      const int orow = b0 + lane;
      const float* xr2 = x + orow * (2 * DD);
      const float ks = kstiff[0];
      const float kd = kdiss[0];
      float g[DD], f[DD];
      float gf = 0.f, gg = 0.f;
#pragma unroll
      for (int i = 0; i < DD; ++i) {
        const float gi = glsP[group][lane][i];
        const float pi = xr2[i];
        const float vi = xr2[DD + i];
        const float fi = ks * (pi - attractor[i]) + kd * vi;
        g[i] = gi; f[i] = fi;
        gf += gi * fi;
        gg += gi * gi;
      }
      const float coef = (gf - qv) / (1.f + gg);
      float* op = out + orow * DD;
#pragma unroll
      for (int i = 0; i < DD; ++i) op[i] = -f[i] + g[i] * coef;
    }
  }
}

extern "C" void kernel_launch(void* const* d_in, const int* in_sizes, int n_in,
                              void* d_out, int out_size, void* d_ws, size_t ws_size,
                              hipStream_t stream) {
  (void)n_in; (void)out_size; (void)d_ws; (void)ws_size;
  const float* x  = (const float*)d_in[0];
  const float* at = (const float*)d_in[1];
  const float* W1 = (const float*)d_in[2];
  const float* b1 = (const float*)d_in[3];
  const float* W2 = (const float*)d_in[4];
  // d_in[5] = b2: unused (only derivatives of h enter acc)
  const float* ks = (const float*)d_in[6];
  const float* kd = (const float*)d_in[7];
  float* out = (float*)d_out;

  const int B = in_sizes[0] / (2 * DD);          // 8192
  const int nblk = B / ROWS_PER_BLOCK;           // 128 blocks x 256 threads (1024 waves)
  hipLaunchKernelGGL(secgeo_wmma_kernel, dim3(nblk), dim3(256), 0, stream,
                     x, at, W1, b1, W2, ks, kd, out);
}